// lossFunc_33638183862599
// MI455X (gfx1250) — compile-verified
//
#include <hip/hip_runtime.h>
#include <hip/hip_bf16.h>
#include <math.h>

#ifndef __has_builtin
#define __has_builtin(x) 0
#endif

#define LC_A 0.5887
#define LC_B 0.2574
#define LC_WEIGHT 0.0001
#define TPB 256
#define VEC 4
#define CHUNK (TPB * VEC)  // 1024 elements per chunk

// ---------------- CDNA5 async global->LDS staging (guarded) ----------------
#if __has_builtin(__builtin_amdgcn_global_load_async_to_lds_b32)
typedef __attribute__((address_space(1))) int* gint_ptr;
typedef __attribute__((address_space(3))) int* lint_ptr;
typedef int v4i_t __attribute__((ext_vector_type(4)));
typedef __attribute__((address_space(1))) v4i_t* gv4_ptr;
typedef __attribute__((address_space(3))) v4i_t* lv4_ptr;
#define HAVE_ASYNC 1
#if __has_builtin(__builtin_amdgcn_global_load_async_to_lds_b128)
// one instruction moves 16B/lane = 512B/wave
__device__ __forceinline__ void async_g2l_16B(const void* gptr, void* lptr) {
  __builtin_amdgcn_global_load_async_to_lds_b128((gv4_ptr)gptr, (lv4_ptr)lptr,
                                                 0, 0);
}
#else
__device__ __forceinline__ void async_g2l_16B(const void* gptr, void* lptr) {
  const char* g = (const char*)gptr;
  char* l = (char*)lptr;
  for (int i = 0; i < 4; ++i)
    __builtin_amdgcn_global_load_async_to_lds_b32((gint_ptr)(g + 4 * i),
                                                  (lint_ptr)(l + 4 * i), 0, 0);
}
#endif
#else
#define HAVE_ASYNC 0
#endif

template <int N>
__device__ __forceinline__ void wait_async() {
#if __has_builtin(__builtin_amdgcn_s_wait_asynccnt)
  __builtin_amdgcn_s_wait_asynccnt((short)N);
#elif HAVE_ASYNC
  asm volatile("s_wait_asynccnt %0" ::"i"(N) : "memory");
#endif
}

// wave32 inclusive scan of small ints
__device__ __forceinline__ int wave_incl_scan(int v, int lane) {
#pragma unroll
  for (int d = 1; d < 32; d <<= 1) {
    int n = __shfl_up(v, d, 32);
    if (lane >= d) v += n;
  }
  return v;
}

// ---------------- kernel: per-row streaming scan + compaction --------------
// One block per row b. Streams pred/labels/scores/problems/first through an
// async-LDS double buffer (2-deep), computes masked BCE partials and the
// stable "zeros2end" rank of each nonzero of pred*lc_mask*first, accumulating
// per-column sums (f64 atomics) and counts (int atomics).
__global__ __launch_bounds__(TPB) void row_scan_kernel(
    const float* __restrict__ pred, const float* __restrict__ labels,
    const int* __restrict__ scores, const int* __restrict__ problems,
    const int* __restrict__ practicing, int n_pract,
    const float* __restrict__ first, int P, double* __restrict__ col_sum,
    int* __restrict__ col_count, double* __restrict__ bce_sum,
    double* __restrict__ cnt_sum) {
  const int tid = threadIdx.x;
  const int b = blockIdx.x;
  const int lane = tid & 31;
  const int wid = tid >> 5;
  const int nwave = TPB / 32;

  __shared__ unsigned pres[32];  // membership bitset for values 0..1023
  __shared__ int s_wtot[TPB / 32];
  __shared__ int s_running;
  __shared__ double s_red[TPB];
#if HAVE_ASYNC
  __shared__ float4 s_px[2][TPB];
  __shared__ float4 s_lb[2][TPB];
  __shared__ int4 s_sc[2][TPB];
  __shared__ int4 s_pr[2][TPB];
  __shared__ float4 s_fi[2][TPB];
#endif

  if (tid < 32) pres[tid] = 0u;
  if (tid == 0) s_running = 0;
  __syncthreads();
  for (int i = tid; i < n_pract; i += TPB) {
    int v = practicing[i];
    if (v >= 0 && v < 1024) atomicOr(&pres[v >> 5], 1u << (v & 31));
  }
  __syncthreads();

  const size_t rb = (size_t)b * (size_t)P;
  const float* gx = pred + rb;  // pred[...,0] is just a flat [B,P]
  const float* gy = labels + rb;
  const int* gs = scores + rb;
  const int* gp = problems + rb;
  const float* gf = first + rb;

  double accB = 0.0;
  int accC = 0;
  const int NCH = P / CHUNK;

#if HAVE_ASYNC
  {
    const int e = tid * VEC;
    async_g2l_16B(gx + e, &s_px[0][tid]);
    async_g2l_16B(gy + e, &s_lb[0][tid]);
    async_g2l_16B(gs + e, &s_sc[0][tid]);
    async_g2l_16B(gp + e, &s_pr[0][tid]);
    async_g2l_16B(gf + e, &s_fi[0][tid]);
  }
#endif

  for (int c = 0; c < NCH; ++c) {
    float4 xv, yv, fv;
    int4 sv, pv4;
#if HAVE_ASYNC
    if (c + 1 < NCH) {
      const int e = (c + 1) * CHUNK + tid * VEC;
      const int nb = (c + 1) & 1;
      async_g2l_16B(gx + e, &s_px[nb][tid]);
      async_g2l_16B(gy + e, &s_lb[nb][tid]);
      async_g2l_16B(gs + e, &s_sc[nb][tid]);
      async_g2l_16B(gp + e, &s_pr[nb][tid]);
      async_g2l_16B(gf + e, &s_fi[nb][tid]);
      wait_async<5>();  // newest 5 may fly; oldest 5 (chunk c) are done
    } else {
      wait_async<0>();
    }
    const int cb = c & 1;
    xv = s_px[cb][tid];
    yv = s_lb[cb][tid];
    sv = s_sc[cb][tid];
    pv4 = s_pr[cb][tid];
    fv = s_fi[cb][tid];
#else
    const int e = c * TPB + tid;  // in float4 units
    xv = ((const float4*)gx)[e];
    yv = ((const float4*)gy)[e];
    sv = ((const int4*)gs)[e];
    pv4 = ((const int4*)gp)[e];
    fv = ((const float4*)gf)[e];
#endif

    const float x[VEC] = {xv.x, xv.y, xv.z, xv.w};
    const float y[VEC] = {yv.x, yv.y, yv.z, yv.w};
    const int sc[VEC] = {sv.x, sv.y, sv.z, sv.w};
    const int pr[VEC] = {pv4.x, pv4.y, pv4.z, pv4.w};
    const float f[VEC] = {fv.x, fv.y, fv.z, fv.w};

    float val[VEC];
    bool nz[VEC];
    int cntL = 0;
#pragma unroll
    for (int k = 0; k < VEC; ++k) {
      // masked BCE-with-logits
      const float m = (sc[k] == 1) ? 1.0f : 0.0f;
      const float bce =
          fmaxf(x[k], 0.0f) - x[k] * y[k] + log1pf(expf(-fabsf(x[k])));
      accB += (double)(bce * m);
      accC += (sc[k] == 1) ? 1 : 0;
      // prediction = pred * lc_mask * first
      const bool member = ((unsigned)pr[k] < 1024u) &&
                          ((pres[pr[k] >> 5] >> (pr[k] & 31)) & 1u);
      val[k] = member ? (x[k] * f[k]) : 0.0f;
      nz[k] = (val[k] != 0.0f);
      cntL += nz[k] ? 1 : 0;
    }

    // ordered rank: elements ordered by p = c*CHUNK + tid*VEC + k
    const int incl = wave_incl_scan(cntL, lane);
    if (lane == 31) s_wtot[wid] = incl;  // wave total
    __syncthreads();
    const int base = s_running;
    int woff = 0;
    for (int w = 0; w < wid; ++w) woff += s_wtot[w];
    int j = base + woff + (incl - cntL);
#pragma unroll
    for (int k = 0; k < VEC; ++k) {
      if (nz[k]) {
        atomicAdd(&col_sum[j], (double)val[k]);
        atomicAdd(&col_count[j], 1);
        ++j;
      }
    }
    __syncthreads();
    if (tid == 0) {
      int tot = 0;
      for (int w = 0; w < nwave; ++w) tot += s_wtot[w];
      s_running = base + tot;
    }
    __syncthreads();
  }

  // block-reduce BCE partial sums (double) and mask count
  s_red[tid] = accB;
  __syncthreads();
  for (int off = TPB / 2; off > 0; off >>= 1) {
    if (tid < off) s_red[tid] += s_red[tid + off];
    __syncthreads();
  }
  if (tid == 0) atomicAdd(bce_sum, s_red[0]);
  __syncthreads();
  s_red[tid] = (double)accC;
  __syncthreads();
  for (int off = TPB / 2; off > 0; off >>= 1) {
    if (tid < off) s_red[tid] += s_red[tid + off];
    __syncthreads();
  }
  if (tid == 0) atomicAdd(cnt_sum, s_red[0]);  // exact: integer-valued doubles
}

// ---------------- kernel: zero the workspace accumulators ------------------
__global__ void init_ws_kernel(double* __restrict__ col_sum,
                               int* __restrict__ col_count,
                               double* __restrict__ bce_sum,
                               double* __restrict__ cnt_sum, int P) {
  int i = blockIdx.x * blockDim.x + threadIdx.x;
  if (i < P) {
    col_sum[i] = 0.0;
    col_count[i] = 0;
  }
  if (i == 0) {
    *bce_sum = 0.0;
    *cnt_sum = 0.0;
  }
}

// ---------------- kernel: finalize -----------------------------------------
__global__ __launch_bounds__(TPB) void finalize_kernel(
    const double* __restrict__ col_sum, const int* __restrict__ col_count,
    const double* __restrict__ bce_sum, const double* __restrict__ cnt_sum,
    int P, float* __restrict__ out) {
  __shared__ double s_red[TPB];
  const int tid = threadIdx.x;
  double acc = 0.0;
  for (int j = tid; j < P; j += TPB) {
    const int c = col_count[j];
    if (c > 0) {
      const double mean = col_sum[j] / (double)c;
      const double fit = LC_A * pow((double)(j + 1), -LC_B);
      const double t = 1.0 - mean - fit;
      acc += t * t;
    }
  }
  s_red[tid] = acc;
  __syncthreads();
  for (int off = TPB / 2; off > 0; off >>= 1) {
    if (tid < off) s_red[tid] += s_red[tid + off];
    __syncthreads();
  }
  if (tid == 0) {
    const double cnt = *cnt_sum;
    const double labeled = (*bce_sum) / cnt / cnt;
    const double total =
        (1.0 - LC_WEIGHT) * labeled + LC_WEIGHT * sqrt(s_red[0]);
    out[0] = (float)total;
  }
}

// ---------------- host launch ----------------------------------------------
extern "C" void kernel_launch(void* const* d_in, const int* in_sizes, int n_in,
                              void* d_out, int out_size, void* d_ws,
                              size_t ws_size, hipStream_t stream) {
  const float* pred = (const float*)d_in[0];
  const float* labels = (const float*)d_in[1];
  const int* scores = (const int*)d_in[2];
  const int* problems = (const int*)d_in[3];
  const int* practicing = (const int*)d_in[4];
  const float* first = (const float*)d_in[5];
  const int n_pract = in_sizes[4];

  const int P = 8192;              // reference constant
  const int Bn = in_sizes[1] / P;  // 2048

  // workspace layout: [P doubles col_sum][bce_sum][cnt_sum][P ints col_count]
  char* ws = (char*)d_ws;
  double* col_sum = (double*)ws;
  double* bce_sum = (double*)(ws + (size_t)P * 8);
  double* cnt_sum = (double*)(ws + (size_t)P * 8 + 8);
  int* col_count = (int*)(ws + (size_t)P * 8 + 16);

  init_ws_kernel<<<(P + TPB - 1) / TPB, TPB, 0, stream>>>(col_sum, col_count,
                                                          bce_sum, cnt_sum, P);
  row_scan_kernel<<<Bn, TPB, 0, stream>>>(pred, labels, scores, problems,
                                          practicing, n_pract, first, P,
                                          col_sum, col_count, bce_sum, cnt_sum);
  finalize_kernel<<<1, TPB, 0, stream>>>(col_sum, col_count, bce_sum, cnt_sum,
                                         P, (float*)d_out);
}